// CaduceusEmbeddingsSTFT_773094113550
// MI455X (gfx1250) — compile-verified
//
#include <hip/hip_runtime.h>
#include <math.h>

#define VOCAB    16
#define D_MODEL  512
#define D_EMB    358
#define D_STFT   154
#define N_FFT    256
#define HOP      64
#define N_FREQ   129
#define BATCH    4
#define SEQLEN   8192
#define N_FRAMES 129
#define N_CH     (N_FREQ * VOCAB)      // 2064 input channels to the projection
#define M_TOTAL  (BATCH * N_FRAMES)    // 516 distinct (b, frame) rows
#define M_PAD    528                   // 33 tiles of 16
#define N_PAD    160                   // 10 tiles of 16 (D_STFT=154 padded)

typedef __attribute__((ext_vector_type(2))) float v2f;
typedef __attribute__((ext_vector_type(8))) float v8f;

// ---------------------------------------------------------------------------
// Prologue: zero-padded, transposed weights  Wt[n][k] = W[k][n] (n<154 else 0)
// plus padded bias. Makes the GEMM inner loop branch-free and K-contiguous
// for both operands.
// ---------------------------------------------------------------------------
__global__ __launch_bounds__(256) void pad_w_kernel(const float* __restrict__ W,
                                                    const float* __restrict__ bias,
                                                    float* __restrict__ Wt,
                                                    float* __restrict__ biasp) {
    const int e = blockIdx.x * 256 + threadIdx.x;     // 0 .. N_PAD*N_CH-1
    const int n = e / N_CH;
    const int k = e - n * N_CH;
    Wt[e] = (n < D_STFT) ? W[(size_t)k * D_STFT + n] : 0.0f;
    if (e < N_PAD) biasp[e] = (e < D_STFT) ? bias[e] : 0.0f;
}

// ---------------------------------------------------------------------------
// Kernel A: per-(b,frame) windowed DFT of the one-hot signal -> |spectrum|.
// One block per frame. feat[frame][freq*16 + v] = |sum_k w[k] e^{-i a} [tok==v]|
// ---------------------------------------------------------------------------
__global__ __launch_bounds__(256) void stft_feat_kernel(const int* __restrict__ ids,
                                                        float* __restrict__ feat) {
    __shared__ float wnd[N_FFT];
    __shared__ float ctab[N_FFT];
    __shared__ float stab[N_FFT];
    __shared__ int   tok[N_FFT];

    const int tid   = threadIdx.x;
    const int frame = blockIdx.x;            // 0..515
    const int b     = frame / N_FRAMES;
    const int f     = frame - b * N_FRAMES;

    {   // one table entry + one token per thread (256 threads == N_FFT)
        const int k = tid;
        const float ang = 6.28318530717958647692f * (float)k * (1.0f / 256.0f);
        const float c = cosf(ang);
        const float s = sinf(ang);
        ctab[k] = c;
        stab[k] = s;
        wnd[k]  = 0.5f - 0.5f * c;           // Hann window
        // reflect-pad index into the original sequence
        int sidx = f * HOP + k - (N_FFT / 2);
        if (sidx < 0)        sidx = -sidx;
        if (sidx >= SEQLEN)  sidx = 2 * (SEQLEN - 1) - sidx;
        tok[k] = ids[b * SEQLEN + sidx];
    }
    __syncthreads();

    for (int p = tid; p < N_CH; p += 256) {
        const int freq = p >> 4;
        const int v    = p & 15;
        float re = 0.0f, im = 0.0f;
        for (int k = 0; k < N_FFT; ++k) {
            const float m = (tok[k] == v) ? wnd[k] : 0.0f;   // cndmask, no branch
            const int   t = (freq * k) & 255;                // angle index mod 2*pi
            re += m * ctab[t];
            im += m * stab[t];
        }
        feat[(size_t)frame * N_CH + p] = sqrtf(re * re + im * im);
    }
}

// ---------------------------------------------------------------------------
// Kernel B: enc[m][n] = feat[m][:] . Wt[n][:] + bias[n] via V_WMMA_F32_16X16X4_F32
// One wave (32 lanes) per 16x16 output tile; K = 2064 = 516 k-steps of 4.
// A layout (ISA): lanes 0-15 hold K={k0,k0+1}, lanes 16-31 hold K={k0+2,k0+3};
// B symmetric in N. Both fragments are contiguous b64 loads -> no divergence.
// ---------------------------------------------------------------------------
__global__ __launch_bounds__(32) void proj_wmma_kernel(const float* __restrict__ feat,
                                                       const float* __restrict__ Wt,
                                                       const float* __restrict__ biasp,
                                                       float* __restrict__ enc) {
    const int mt   = blockIdx.x;         // 0..32
    const int nt   = blockIdx.y;         // 0..9
    const int lane = threadIdx.x;        // 0..31
    const int half = lane >> 4;          // 0 or 1
    const int lid  = lane & 15;

    const int m = mt * 16 + lid;         // A-matrix row for this lane (< M_PAD)
    const int n = nt * 16 + lid;         // B-matrix col for this lane (< N_PAD)

    const float* arow = feat + (size_t)m * N_CH + half * 2;
    const float* brow = Wt   + (size_t)n * N_CH + half * 2;

    v8f acc = {};
    #pragma unroll 8
    for (int k0 = 0; k0 < N_CH; k0 += 4) {
        const v2f a = *(const v2f*)(arow + k0);
        const v2f b = *(const v2f*)(brow + k0);
        acc = __builtin_amdgcn_wmma_f32_16x16x4_f32(
            /*neg_a=*/false, a, /*neg_b=*/false, b,
            /*c_mod=*/(short)0, acc, /*reuse_a=*/false, /*reuse_b=*/false);
    }

    const float bval = biasp[n];
    // C/D layout: VGPR r -> M = r + 8*half, N = lid
    #pragma unroll
    for (int r = 0; r < 8; ++r) {
        const int mm = mt * 16 + r + half * 8;
        if (mm < M_TOTAL && n < D_STFT) {
            enc[(size_t)mm * D_STFT + n] = acc[r] + bval;
        }
    }
}

// ---------------------------------------------------------------------------
// Kernel C: streaming assemble of the 67 MB output.
// out[b][l][0:358]   = emb_table[input_ids[b][l]]
// out[b][l][358:512] = enc[b*129 + (l*129 >> 13)]
// One block per (b,l); 256 threads x float2 = 512 floats.
// ---------------------------------------------------------------------------
__global__ __launch_bounds__(256) void assemble_kernel(const int* __restrict__ ids,
                                                       const float* __restrict__ emb,
                                                       const float* __restrict__ enc,
                                                       float* __restrict__ out) {
    const int pos = blockIdx.x;          // b*SEQLEN + l
    const int b   = pos >> 13;
    const int l   = pos & (SEQLEN - 1);
    const int d   = threadIdx.x * 2;

    const int id = ids[pos];
    const int f  = (l * N_FRAMES) >> 13; // l * 129 / 8192

    float2 val;
    if (d < D_EMB) {
        const float* src = emb + (size_t)id * D_EMB + d;
        val.x = src[0];
        val.y = src[1];
    } else {
        const int j = d - D_EMB;         // 0..152, D_STFT even
        const float* src = enc + (size_t)(b * N_FRAMES + f) * D_STFT + j;
        val.x = src[0];
        val.y = src[1];
    }
    *(float2*)(out + (size_t)pos * D_MODEL + d) = val;
}

// ---------------------------------------------------------------------------
extern "C" void kernel_launch(void* const* d_in, const int* in_sizes, int n_in,
                              void* d_out, int out_size, void* d_ws, size_t ws_size,
                              hipStream_t stream) {
    const int*   ids  = (const int*)d_in[0];     // (4, 8192) int32
    const float* emb  = (const float*)d_in[1];   // (16, 358)
    const float* W    = (const float*)d_in[2];   // (2064, 154)
    const float* bias = (const float*)d_in[3];   // (154,)
    float* out = (float*)d_out;

    float* feat  = (float*)d_ws;                           // M_PAD * N_CH floats
    float* Wt    = feat  + (size_t)M_PAD * N_CH;           // N_PAD * N_CH floats
    float* biasp = Wt    + (size_t)N_PAD * N_CH;           // N_PAD floats
    float* enc   = biasp + N_PAD;                          // M_TOTAL * D_STFT floats

    hipLaunchKernelGGL(pad_w_kernel, dim3((N_PAD * N_CH) / 256), dim3(256), 0, stream,
                       W, bias, Wt, biasp);
    hipLaunchKernelGGL(stft_feat_kernel, dim3(M_TOTAL), dim3(256), 0, stream, ids, feat);
    hipLaunchKernelGGL(proj_wmma_kernel, dim3(M_PAD / 16, 10), dim3(32), 0, stream,
                       feat, Wt, biasp, enc);
    hipLaunchKernelGGL(assemble_kernel, dim3(BATCH * SEQLEN), dim3(256), 0, stream,
                       ids, emb, enc, out);
}